// LocalGrouper_71760313581928
// MI455X (gfx1250) — compile-verified
//
#include <hip/hip_runtime.h>
#include <hip/hip_bf16.h>

// Problem constants (from reference setup_inputs / module constants)
#define B 8
#define N 16384
#define D 64
#define S 512        // GROUPS
#define K 32         // kneighbors

typedef __attribute__((ext_vector_type(2))) float v2f;
typedef __attribute__((ext_vector_type(8))) float v8f;

// ---------------------------------------------------------------------------
// Kernel 1: Farthest Point Sampling. One block per batch; dist[] lives in
// global scratch (B*N floats). Sequential 512-step loop with block-wide
// argmax (first-index tie-break to match jnp.argmax).
// ---------------------------------------------------------------------------
#define FPS_THREADS 512

__global__ __launch_bounds__(FPS_THREADS)
void fps_kernel(const float* __restrict__ xyz, int* __restrict__ sidx,
                float* __restrict__ sxyz, float* __restrict__ snorm,
                float* __restrict__ dist_ws) {
  const int b    = blockIdx.x;
  const int tid  = threadIdx.x;
  const int lane = tid & 31;
  const int wid  = tid >> 5;
  const int NW   = FPS_THREADS / 32;

  const float* xb   = xyz + (size_t)b * 3 * N;   // x row, y at +N, z at +2N
  float*       dist = dist_ws + (size_t)b * N;

  __shared__ float cent[3];
  __shared__ int   curfar;
  __shared__ float rd[NW];
  __shared__ int   ri[NW];

  for (int n = tid; n < N; n += FPS_THREADS) dist[n] = 1e10f;
  if (tid == 0) curfar = 0;
  __syncthreads();

  for (int it = 0; it < S; ++it) {
    const int far = curfar;
    if (tid == 0) {
      float cx = xb[far], cy = xb[N + far], cz = xb[2 * N + far];
      cent[0] = cx; cent[1] = cy; cent[2] = cz;
      sidx[b * S + it] = far;
      sxyz[((size_t)b * S + it) * 3 + 0] = cx;
      sxyz[((size_t)b * S + it) * 3 + 1] = cy;
      sxyz[((size_t)b * S + it) * 3 + 2] = cz;
      snorm[b * S + it] = cx * cx + cy * cy + cz * cz;
    }
    __syncthreads();
    const float cx = cent[0], cy = cent[1], cz = cent[2];

    float bestd = -1.0f;
    int   besti = 0x7fffffff;
    for (int n = tid; n < N; n += FPS_THREADS) {
      float dx = xb[n] - cx, dy = xb[N + n] - cy, dz = xb[2 * N + n] - cz;
      float d  = dx * dx + dy * dy + dz * dz;
      float nd = fminf(dist[n], d);
      dist[n]  = nd;
      if (nd > bestd) { bestd = nd; besti = n; }  // strict > keeps smallest n
    }
    // wave32 reduction, tie-break toward smaller index
    for (int off = 16; off > 0; off >>= 1) {
      float od = __shfl_down(bestd, off);
      int   oi = __shfl_down(besti, off);
      if (od > bestd || (od == bestd && oi < besti)) { bestd = od; besti = oi; }
    }
    if (lane == 0) { rd[wid] = bestd; ri[wid] = besti; }
    __syncthreads();
    if (wid == 0) {
      bestd = (lane < NW) ? rd[lane] : -1.0f;
      besti = (lane < NW) ? ri[lane] : 0x7fffffff;
      for (int off = 16; off > 0; off >>= 1) {
        float od = __shfl_down(bestd, off);
        int   oi = __shfl_down(besti, off);
        if (od > bestd || (od == bestd && oi < besti)) { bestd = od; besti = oi; }
      }
      if (lane == 0) curfar = besti;
    }
    __syncthreads();
  }
}

// ---------------------------------------------------------------------------
// Kernel 2: fused distance (WMMA f32 16x16x4) + top-K selection.
// One BLOCK (8 waves) owns a 16-row tile of sampled points; the 8 waves split
// the N candidates (2048 each). Per 16-candidate tile:
//   dot = A(16x4) x B(4x16) via v_wmma_f32_16x16x4_f32  (K=4: xyz + 0 pad)
//   d2  = |a|^2 + |b|^2 - 2*dot
// Each wave keeps an UNSORTED per-row top-32 via max-replacement (fixed-bound
// branchless rescan -> no EXEC-stack cascades). Final 8-way merge (256 -> 32,
// (d, idx)-lexicographic) reproduces lax.top_k's sorted/stable order.
// ---------------------------------------------------------------------------
#define KNN_WAVES 8
#define NT        (N / 16)               // 1024 candidate tiles
#define NT_PER_W  (NT / KNN_WAVES)       // 128 tiles per wave
#define LISTW     (KNN_WAVES * K)        // 256 merged entries per row

__global__ __launch_bounds__(KNN_WAVES * 32)
void knn_kernel(const float* __restrict__ xyz, const float* __restrict__ sxyz,
                const float* __restrict__ snorm, int* __restrict__ knn) {
  __shared__ float a2s[16];
  __shared__ float tile[KNN_WAVES][16][16];
  __shared__ float Ld[16][LISTW];
  __shared__ int   Li[16][LISTW];

  const int tid  = threadIdx.x;
  const int lane = tid & 31;
  const int w    = tid >> 5;
  const int b    = blockIdx.x / (S / 16);
  const int st   = blockIdx.x % (S / 16);
  const int s0   = st * 16;
  const int m    = lane & 15;
  const int hi   = lane >> 4;

  const float* xb = xyz + (size_t)b * 3 * N;

  // A matrix: 16 sampled points x 4 (coords padded with 0). Same for all waves.
  // ISA layout: lanes 0-15 hold K=0 (VGPR0) / K=1 (VGPR1); lanes 16-31 K=2 / K=3.
  const float* sp = sxyz + ((size_t)b * S + s0 + m) * 3;
  v2f a;
  a.x = hi ? sp[2] : sp[0];
  a.y = hi ? 0.0f  : sp[1];

  if (w == 0 && !hi) a2s[m] = snorm[b * S + s0 + m];
  // init merged list storage
  {
    float* ldf = &Ld[0][0];
    int*   lif = &Li[0][0];
    for (int e = tid; e < 16 * LISTW; e += KNN_WAVES * 32) {
      ldf[e] = INFINITY; lif[e] = 0x7fffffff;
    }
  }
  __syncthreads();

  // per (row, wave) unsorted top-K state (meaningful for lanes 0-15)
  float maxv = INFINITY;   // current max of my 32-slot slice
  int   maxp = 0;          // its position

  const int nt0 = w * NT_PER_W;
  for (int t = 0; t < NT_PER_W; ++t) {
    const int nt = nt0 + t;
    const int n0 = nt * 16;
    const int n  = n0 + m;
    const float px = xb[n], py = xb[N + n], pz = xb[2 * N + n];
    if (t + 1 < NT_PER_W) {                      // prefetch next candidate tile
      __builtin_prefetch(xb + n + 16, 0, 3);
      __builtin_prefetch(xb + N + n + 16, 0, 3);
      __builtin_prefetch(xb + 2 * N + n + 16, 0, 3);
    }
    const float b2 = px * px + py * py + pz * pz;

    // B matrix: 4 x 16 candidates, same K striping as A.
    v2f bm;
    bm.x = hi ? pz   : px;
    bm.y = hi ? 0.0f : py;

    v8f c = {};
    c = __builtin_amdgcn_wmma_f32_16x16x4_f32(
        /*neg_a=*/false, a, /*neg_b=*/false, bm,
        /*c_mod=*/(short)0, c, /*reuse_a=*/false, /*reuse_b=*/false);

    // C/D layout: VGPR r holds row (r + 8*hi), column = lane&15.
#pragma unroll
    for (int r = 0; r < 8; ++r) {
      const int row = r + 8 * hi;
      tile[w][row][m] = a2s[row] + b2 - 2.0f * c[r];
    }
    __syncthreads();   // uniform trip count across all 8 waves

    if (!hi) {                                   // lane m owns row m
      float* Lrow = &Ld[m][w * K];
      int*   Irow = &Li[m][w * K];
      for (int j = 0; j < 16; ++j) {
        const float d = tile[w][m][j];
        if (d < maxv) {                          // replace current max slot
          Lrow[maxp] = d;
          Irow[maxp] = n0 + j;
          // branchless fixed-bound rescan for new max (first-max on ties)
          float mx = -INFINITY; int mp = 0;
#pragma unroll
          for (int k = 0; k < K; ++k) {
            float v = Lrow[k];
            bool gt = v > mx;
            mx = gt ? v : mx;
            mp = gt ? k : mp;
          }
          maxv = mx; maxp = mp;
        }
      }
    }
    __syncthreads();
  }

  // Parallel merge: 16 rows spread over 8 waves (2 rows/wave, lanes 0 & 16).
  const int row = w * 2 + hi;
  if (m == 0) {
    int* out = knn + ((size_t)b * S + s0 + row) * K;
    for (int k2 = 0; k2 < K; ++k2) {
      float best = INFINITY; int besti = 0x7fffffff; int bestp = 0;
      for (int e = 0; e < LISTW; ++e) {
        float v  = Ld[row][e];
        int   vi = Li[row][e];
        bool take = (v < best) || (v == best && vi < besti);
        best  = take ? v  : best;
        besti = take ? vi : besti;
        bestp = take ? e  : bestp;
      }
      out[k2] = besti;
      Ld[row][bestp] = INFINITY;                 // remove selected entry
    }
  }
}

// ---------------------------------------------------------------------------
// Kernel 3: gather features/xyz at knn indices, center xyz, write outputs.
// d_out = [ new_xyz (B*3*S) | new_features (B*(D+3)*S*K) ].
// ---------------------------------------------------------------------------
__global__ __launch_bounds__(256)
void gather_kernel(const float* __restrict__ xyz, const float* __restrict__ feat,
                   const int* __restrict__ knn, const float* __restrict__ sxyz,
                   float* __restrict__ out) {
  const int b   = blockIdx.x / S;
  const int s   = blockIdx.x % S;
  const int tid = threadIdx.x;

  __shared__ int   kid[K];
  __shared__ float cen[3];
  if (tid < K) kid[tid] = knn[((size_t)b * S + s) * K + tid];
  if (tid < 3) cen[tid] = sxyz[((size_t)b * S + s) * 3 + tid];
  __syncthreads();

  float* newxyz = out;                         // [B,3,S]
  float* nf     = out + (size_t)B * 3 * S;     // [B,D+3,S,K]
  if (tid < 3) newxyz[((size_t)b * 3 + tid) * S + s] = cen[tid];

  const int C = D + 3;
  for (int e = tid; e < C * K; e += blockDim.x) {
    const int ch  = e >> 5;        // / K
    const int k   = e & (K - 1);
    const int idx = kid[k];
    float v;
    if (ch < D) v = feat[((size_t)b * D + ch) * N + idx];
    else        v = xyz[((size_t)b * 3 + (ch - D)) * N + idx] - cen[ch - D];
    nf[(((size_t)b * C + ch) * S + s) * K + k] = v;
  }
}

// ---------------------------------------------------------------------------
extern "C" void kernel_launch(void* const* d_in, const int* in_sizes, int n_in,
                              void* d_out, int out_size, void* d_ws, size_t ws_size,
                              hipStream_t stream) {
  const float* xyz  = (const float*)d_in[0];   // [B,3,N] f32
  const float* feat = (const float*)d_in[1];   // [B,D,N] f32

  char* ws = (char*)d_ws;
  size_t off = 0;
  int*   sidx  = (int*)  (ws + off); off += (size_t)B * S * sizeof(int);        // sampled indices
  float* sxyz  = (float*)(ws + off); off += (size_t)B * S * 3 * sizeof(float);  // sampled coords [B,S,3]
  float* snorm = (float*)(ws + off); off += (size_t)B * S * sizeof(float);      // |sample|^2
  int*   knn   = (int*)  (ws + off); off += (size_t)B * S * K * sizeof(int);    // knn indices
  float* dist  = (float*)(ws + off); off += (size_t)B * N * sizeof(float);      // FPS running min dist

  fps_kernel<<<B, FPS_THREADS, 0, stream>>>(xyz, sidx, sxyz, snorm, dist);
  knn_kernel<<<B * (S / 16), KNN_WAVES * 32, 0, stream>>>(xyz, sxyz, snorm, knn);
  gather_kernel<<<B * S, 256, 0, stream>>>(xyz, feat, knn, sxyz, (float*)d_out);
}